// ModeModeWavelengthDependentDiffractionLayer_38208029065724
// MI455X (gfx1250) — compile-verified
//
#include <hip/hip_runtime.h>
#include <math.h>

// ---------------------------------------------------------------------------
// MI455X (gfx1250) wavelength-dependent diffraction layer.
//   x' = (xr + i xi) * exp(i * phase_mask * mult)
//   out = ifft2( fft2(x') * TF(c) ), energy-renormalized
// 1024-pt FFT = four-step: 1024 = 32x32, radix-32 stages as complex 32x32x32
// matmuls on V_WMMA_F32_16X16X4_F32 (wave32 matrix pipe).  Global->LDS tile
// staging uses the Tensor Data Mover (tensor_load_to_lds) with hardware
// padding that reproduces the bank-conflict-free PAD=33 layout.
// ---------------------------------------------------------------------------

typedef __attribute__((ext_vector_type(2))) float v2f;
typedef __attribute__((ext_vector_type(8))) float v8f;

#define UNITS   1024
#define CW      3
#define NB      18            // B*C = 6*3
#define HWSZ    (1024*1024)
#define PAD     33            // padded row stride (float2)
#define TWO_PI_F 6.28318530717958647692f

__device__ __forceinline__ float ffreq(int j) {
  // jnp.fft.fftfreq(1024, d=8e-6):  (j<512 ? j : j-1024) / (1024*8e-6)
  return (float)(j < 512 ? j : j - 1024) * 122.0703125f;
}

// ---------------------------------------------------------------------------
// Tensor Data Mover staging: load a 1024-element (float2) line into LDS with
// one pad element every 32 (PAD=33 layout).  strideElems = 32 -> contiguous
// row tile (32x32); strideElems = 1024 -> strided column gather (1x1024).
// Descriptor packing per CDNA5 ISA section 8.3/8.4.  This toolchain exposes
// the 6-arg builtin: (v4u g0, v8i g1, v4i g2, v4i g3, v8i g4, i32 cpol).
// ---------------------------------------------------------------------------
#if __has_builtin(__builtin_amdgcn_tensor_load_to_lds)
#define HAVE_TDM 1
typedef __attribute__((ext_vector_type(4))) unsigned tdm_g0_t;
typedef __attribute__((ext_vector_type(8))) int      tdm_g1_t;
typedef __attribute__((ext_vector_type(4))) int      tdm_g2_t;

__device__ __forceinline__ unsigned lds_addr_of(void* p) {
  return (unsigned)(uintptr_t)(__attribute__((address_space(3))) char*)p;
}

__device__ __forceinline__ void tdm_load_line(const float2* gsrc, unsigned ldsOff,
                                              unsigned strideElems)
{
  unsigned long long ga = (unsigned long long)(uintptr_t)gsrc;
  tdm_g0_t g0;
  g0[0] = 1u;                                                  // count=1 (valid)
  g0[1] = (unsigned)__builtin_amdgcn_readfirstlane((int)ldsOff);
  g0[2] = (unsigned)__builtin_amdgcn_readfirstlane((int)(unsigned)ga);
  g0[3] = (unsigned)__builtin_amdgcn_readfirstlane((int)((unsigned)(ga >> 32) | (2u << 30))); // type=2

  const unsigned tile0 = (strideElems == 32u) ? 32u : 1u;
  const unsigned tile1 = (strideElems == 32u) ? 32u : 1024u;
  const unsigned td0   = tile0;                                 // tensor_dim0
  const unsigned td1   = 1024u;                                 // tensor_dim1
  tdm_g1_t g1;
  // data_size=3 (8B), pad_enable, pad_interval=5 (32 elems), pad_amount=1 (2 dwords)
  g1[0] = (int)(0x00030000u | (1u << 20) | (5u << 22) | (1u << 25));
  g1[1] = (int)((td0 & 0xFFFFu) << 16);                         // [47:32]=atomic addr 0
  g1[2] = (int)((td0 >> 16) | ((td1 & 0xFFFFu) << 16));
  g1[3] = (int)((td1 >> 16) | (tile0 << 16));
  g1[4] = (int)tile1;                                           // tile_dim2 = 0
  g1[5] = (int)strideElems;                                     // tensor_dim0_stride lo
  g1[6] = 0;                                                    // stride0 hi | stride1 lo
  g1[7] = 0;                                                    // stride1 hi
  tdm_g2_t gz4 = {0, 0, 0, 0};
  tdm_g1_t gz8 = {0, 0, 0, 0, 0, 0, 0, 0};
  __builtin_amdgcn_tensor_load_to_lds(g0, g1, gz4, gz4, gz8, 0);
}
#endif

// ---------------------------------------------------------------------------
// Complex 32x32 * 32x32 matmul on WMMA f32 16x16x4, one wave.
//   D = A * B   (optionally multiplied by twiddle tw[(m*n)&1023]^tws)
// A: A[m*lda+k] (float2), B: B[k*ldb+n], D: D[m*ldd+n].
// ISA 7.12.2 layouts: A(16x4 f32): lanes 0-15 K={0,1}, lanes 16-31 K={2,3};
// C/D tile: VGPR v holds m = v + 8*(lane>=16), n = lane&15.
// f32 WMMA has no A/B neg modifiers -> VALU-negate Ai for the -Ai*Bi term.
// ---------------------------------------------------------------------------
__device__ __forceinline__ void cmatmul32(const float2* __restrict__ A, int lda,
                                          const float2* __restrict__ B, int ldb,
                                          float2* __restrict__ D, int ldd,
                                          int lane,
                                          const float2* __restrict__ tw, float tws)
{
  const int half = lane >> 4;    // 0 or 1
  const int l15  = lane & 15;
  #pragma unroll
  for (int tm = 0; tm < 32; tm += 16) {
    #pragma unroll
    for (int tn = 0; tn < 32; tn += 16) {
      v8f accr = {};
      v8f acci = {};
      const int mA = tm + l15;
      const int nB = tn + l15;
      #pragma unroll
      for (int kb = 0; kb < 32; kb += 4) {
        const int k0 = kb + 2 * half;
        float2 a0 = A[mA * lda + k0];
        float2 a1 = A[mA * lda + k0 + 1];
        float2 b0 = B[k0 * ldb + nB];
        float2 b1 = B[(k0 + 1) * ldb + nB];
        v2f Ar = {a0.x, a1.x};
        v2f Ai = {a0.y, a1.y};
        v2f Br = {b0.x, b1.x};
        v2f Bi = {b0.y, b1.y};
        v2f nAi = -Ai;
        // Cr += Ar*Br + (-Ai)*Bi ; Ci += Ar*Bi + Ai*Br
        accr = __builtin_amdgcn_wmma_f32_16x16x4_f32(false, Ar,  false, Br, (short)0, accr, false, false);
        accr = __builtin_amdgcn_wmma_f32_16x16x4_f32(false, nAi, false, Bi, (short)0, accr, false, false);
        acci = __builtin_amdgcn_wmma_f32_16x16x4_f32(false, Ar,  false, Bi, (short)0, acci, false, false);
        acci = __builtin_amdgcn_wmma_f32_16x16x4_f32(false, Ai,  false, Br, (short)0, acci, false, false);
      }
      const int mBase = tm + 8 * half;
      #pragma unroll
      for (int v = 0; v < 8; ++v) {
        const int m = mBase + v;
        float re = accr[v], im = acci[v];
        if (tw) {
          float2 t = tw[(m * nB) & 1023];
          float cs = t.x, sn = t.y * tws;   // tws=-1 -> conjugate (inverse pass)
          float r2 = re * cs - im * sn;
          im = re * sn + im * cs;
          re = r2;
        }
        D[m * ldd + nB] = make_float2(re, im);
      }
    }
  }
}

// ---------------------------------------------------------------------------
// 1024-pt FFT of one line in `buf` (Xrc layout: buf[(e>>5)*PAD+(e&31)] =
// line[e]).  Result: line[k] = buf[(k&31)*PAD + (k>>5)].
// ---------------------------------------------------------------------------
__device__ __forceinline__ void fft1024_inplace(float2* buf, float2* scratch,
                                                const float2* Fm,
                                                const float2* tw, float tws,
                                                int lane)
{
  __syncthreads();
  cmatmul32(Fm, 32, buf, PAD, scratch, PAD, lane, tw, tws);  // T1 = F*Xrc (+twiddle)
  __syncthreads();
  cmatmul32(scratch, PAD, Fm, 32, buf, PAD, lane, (const float2*)0, 0.f); // T2 = T1*F
  __syncthreads();
}

__device__ __forceinline__ void buildF(float2* Fm, float s, int tid) {
  for (int e = tid; e < 1024; e += 64) {
    int a = e >> 5, b = e & 31;
    float th = s * (TWO_PI_F / 32.f) * (float)((a * b) & 31);
    float sn, cs;
    __sincosf(th, &sn, &cs);
    Fm[e] = make_float2(cs, sn);
  }
}

__device__ __forceinline__ void buildTw(float2* tw, float s, int tid) {
  for (int e = tid; e < 1024; e += 64) {
    float th = s * (TWO_PI_F / 1024.f) * (float)e;
    float sn, cs;
    __sincosf(th, &sn, &cs);
    tw[e] = make_float2(cs, sn);
  }
}

// ---------------------------------------------------------------------------
// K1: modulation  x' = (xr+i xi)*exp(i*pm*mult);  e_in partial sums -> red[bc]
// ---------------------------------------------------------------------------
__global__ void __launch_bounds__(256) modulate_kernel(
    const float* __restrict__ xr, const float* __restrict__ xi,
    const float* __restrict__ pm, const float* __restrict__ wc,
    float2* __restrict__ xbuf, float* __restrict__ red)
{
  const int idx = blockIdx.x * 256 + threadIdx.x;
  const int bc  = idx / HWSZ;            // HWSZ % 256 == 0 -> uniform per block
  const int b   = bc / CW;
  const int c   = bc - b * CW;
  // coeff forced to 1.0 at base wavelength idx 1, clip(.,0.5,1.5)*DEPTH_FACTOR(=1)
  float coeff = (c == 1) ? 1.0f : fminf(fmaxf(wc[bc], 0.5f), 1.5f);
  float ph = pm[b * HWSZ + (idx - bc * HWSZ)] * coeff;
  float sn, cs;
  __sincosf(ph, &sn, &cs);
  float a = xr[idx], bb = xi[idx];
  xbuf[idx] = make_float2(a * cs - bb * sn, a * sn + bb * cs);

  // e_in: |x'|^2 == xr^2 + xi^2 (phase rotation preserves magnitude)
  float e = a * a + bb * bb;
  __shared__ float sred[256];
  sred[threadIdx.x] = e;
  __syncthreads();
  for (int sN = 128; sN > 0; sN >>= 1) {
    if (threadIdx.x < sN) sred[threadIdx.x] += sred[threadIdx.x + sN];
    __syncthreads();
  }
  if (threadIdx.x == 0) atomicAdd(&red[bc], sred[0]);
}

// ---------------------------------------------------------------------------
// K2/K4: row FFT pass (one wave per row, 2 waves/block, TDM-staged tiles).
// ---------------------------------------------------------------------------
__global__ void __launch_bounds__(64) fft_rows_kernel(
    float2* __restrict__ xbuf, float* __restrict__ red,
    float s, float scale, int doE)
{
  __shared__ float2 Fm[1024];
  __shared__ float2 Tw[1024];
  __shared__ float2 work[2][2][PAD * 32];
  const int tid = threadIdx.x, lane = tid & 31, wv = tid >> 5;

  const int gw = blockIdx.x * 2 + wv;         // global line id: 0..18431
  const int bc = gw >> 10;
  float2* g  = xbuf + (size_t)gw * UNITS;
  float2* bA = work[wv][0];
  float2* bB = work[wv][1];

#if defined(HAVE_TDM)
  tdm_load_line(g, lds_addr_of(bA), 32u);     // async DMA while tables build
#endif
  buildF(Fm, s, tid);
  buildTw(Tw, s, tid);
#if defined(HAVE_TDM)
  __builtin_amdgcn_s_wait_tensorcnt(0);
#else
  for (int j = 0; j < 32; ++j)
    bA[j * PAD + lane] = g[j * 32 + lane];
#endif
  __syncthreads();

  fft1024_inplace(bA, bB, Fm, Tw, 1.f, lane);

  float esum = 0.f;
  for (int j = 0; j < 32; ++j) {
    float2 v = bA[lane * PAD + j];            // line[k], k = j*32+lane
    v.x *= scale; v.y *= scale;
    g[j * 32 + lane] = v;
    esum += v.x * v.x + v.y * v.y;
  }
  if (doE) {
    for (int off = 16; off > 0; off >>= 1) esum += __shfl_xor(esum, off, 32);
    if (lane == 0) atomicAdd(&red[NB + bc], esum);
  }
}

// ---------------------------------------------------------------------------
// K3: fused column pass: fwd col FFT -> angular-spectrum TF -> inv col FFT.
// TDM performs the strided (8KB-stride) column gather; working set is
// L2-resident (144MB < 192MB global L2).
// ---------------------------------------------------------------------------
__global__ void __launch_bounds__(64) fft_cols_tf_kernel(float2* __restrict__ xbuf)
{
  __shared__ float2 Ff[1024];   // forward table (s=-1)
  __shared__ float2 Fi[1024];   // inverse table (s=+1)
  __shared__ float2 Tw[1024];   // forward twiddles (conjugated for inverse)
  __shared__ float2 work[2][2][PAD * 32];
  const int tid = threadIdx.x, lane = tid & 31, wv = tid >> 5;

  const int gw  = blockIdx.x * 2 + wv;
  const int bc  = gw >> 10;
  const int col = gw & 1023;
  const int c   = bc % CW;
  float2* img = xbuf + (size_t)bc * HWSZ;
  float2* bA  = work[wv][0];
  float2* bB  = work[wv][1];

#if defined(HAVE_TDM)
  tdm_load_line(img + col, lds_addr_of(bA), 1024u);  // strided column gather
#endif
  buildF(Ff, -1.f, tid);
  buildF(Fi, +1.f, tid);
  buildTw(Tw, -1.f, tid);
#if defined(HAVE_TDM)
  __builtin_amdgcn_s_wait_tensorcnt(0);
#else
  for (int j = 0; j < 32; ++j)
    bA[j * PAD + lane] = img[(size_t)(j * 32 + lane) * UNITS + col];
#endif
  __syncthreads();

  fft1024_inplace(bA, bB, Ff, Tw, 1.f, lane);        // forward along H

  // TF: arg = 1/lam^2 - (fy^2+fx^2); tf = exp(i*2*pi*z*sqrt(arg)) or 0
  const float lam[CW] = {4.5e-7f, 5.32e-7f, 6.33e-7f};
  const float il2 = 1.f / (lam[c] * lam[c]);
  const float fxv = ffreq(col);
  const float fx2 = fxv * fxv;
  for (int j = 0; j < 32; ++j) {
    const int k = j * 32 + lane;
    float2 v = bA[lane * PAD + j];                   // spectrum line[k]
    float fyv = ffreq(k);
    float arg = il2 - (fyv * fyv + fx2);
    float2 t;
    if (arg >= 0.f) {
      float kz = TWO_PI_F * 0.05f * sqrtf(arg);
      t = make_float2(cosf(kz), sinf(kz));
    } else {
      t = make_float2(0.f, 0.f);
    }
    // multiply + repack into Xrc layout for the inverse transform
    bB[j * PAD + lane] = make_float2(v.x * t.x - v.y * t.y,
                                     v.x * t.y + v.y * t.x);
  }

  fft1024_inplace(bB, bA, Fi, Tw, -1.f, lane);       // inverse along H (conj tw)

  const float scale = 1.f / 1024.f;                  // half of the ifft2 1/N^2
  for (int j = 0; j < 32; ++j) {
    float2 v = bB[lane * PAD + j];
    v.x *= scale; v.y *= scale;
    img[(size_t)(j * 32 + lane) * UNITS + col] = v;
  }
}

// ---------------------------------------------------------------------------
// K5: energy renormalization, write interleaved complex64 to d_out.
// ---------------------------------------------------------------------------
__global__ void __launch_bounds__(256) finalize_kernel(
    const float2* __restrict__ xbuf, const float* __restrict__ red,
    float* __restrict__ out)
{
  const int idx = blockIdx.x * 256 + threadIdx.x;
  const int bc  = idx / HWSZ;
  const float inv_hw = 1.f / (float)HWSZ;
  float ein  = red[bc]      * inv_hw;
  float eout = red[NB + bc] * inv_hw;
  float ratio = fminf(fmaxf(sqrtf(ein / eout), 0.5f), 2.0f);
  float sR = (eout > 1e-8f) ? ratio : 1.0f;
  float2 v = xbuf[idx];
  out[2 * idx]     = v.x * sR;
  out[2 * idx + 1] = v.y * sR;
}

// ---------------------------------------------------------------------------
extern "C" void kernel_launch(void* const* d_in, const int* in_sizes, int n_in,
                              void* d_out, int out_size, void* d_ws, size_t ws_size,
                              hipStream_t stream) {
  const float* xr = (const float*)d_in[0];   // [6,3,1024,1024] f32
  const float* xi = (const float*)d_in[1];   // [6,3,1024,1024] f32
  const float* pm = (const float*)d_in[2];   // [6,1024,1024]   f32
  const float* wc = (const float*)d_in[3];   // [6,3]           f32

  float*  red  = (float*)d_ws;                      // 36 reduction slots
  float2* xbuf = (float2*)((char*)d_ws + 256);      // 18M complex64 field

  const int total = in_sizes[0];                    // 18,874,368
  const int lines = total / UNITS;                  // 18,432

  (void)hipMemsetAsync(d_ws, 0, 256, stream);       // zero e_in/e_out sums

  modulate_kernel<<<total / 256, 256, 0, stream>>>(xr, xi, pm, wc, xbuf, red);
  fft_rows_kernel<<<lines / 2, 64, 0, stream>>>(xbuf, red, -1.f, 1.f, 0);
  fft_cols_tf_kernel<<<lines / 2, 64, 0, stream>>>(xbuf);
  fft_rows_kernel<<<lines / 2, 64, 0, stream>>>(xbuf, red, +1.f, 1.f / 1024.f, 1);
  finalize_kernel<<<total / 256, 256, 0, stream>>>(xbuf, red, (float*)d_out);
}